// Transformer_11149735100720
// MI455X (gfx1250) — compile-verified
//
#include <hip/hip_runtime.h>
#include <stdint.h>

#define RES    64
#define IMG_W  256
#define IMG_H  256
#define TILE   16
#define PI_F   3.14159265358979323846f

// ---- CDNA5 async global->LDS path -------------------------------------------
// Probe-verified: __builtin_amdgcn_global_load_async_to_lds_b128 takes
// (v4i32 addrspace(1)* src, v4i32 addrspace(3)* dst, imm offset, imm cpol).
typedef int v4i __attribute__((ext_vector_type(4)));
typedef __attribute__((address_space(1))) v4i gv4i_t;   // global v4i32
typedef __attribute__((address_space(3))) v4i lv4i_t;   // LDS v4i32

#if defined(__gfx1250__) && __has_builtin(__builtin_amdgcn_global_load_async_to_lds_b128)
#define USE_ASYNC_LDS 1
#else
#define USE_ASYNC_LDS 0
#endif

__device__ __forceinline__ void wait_asynccnt0() {
#if defined(__HIP_DEVICE_COMPILE__)
#if __has_builtin(__builtin_amdgcn_s_wait_asynccnt)
    __builtin_amdgcn_s_wait_asynccnt(0);
#else
    asm volatile("s_wait_asynccnt 0" ::: "memory");
#endif
#endif
}

// ----------------------------------------------------------------------------
// One block per batch item. 256 threads (8 wave32).
//   Stage 1: lanes 0..63 compute radius(col), cos(row), sin(row) into LDS and
//            issue 64 async B128 copies staging a 16x16 image tile into LDS
//            (tracked on ASYNCcnt, drained with s_wait_asynccnt).
//   Stage 2: each thread emits 4 groups of 4 consecutive pixels; per group one
//            ds_load_b128 of radii, shared cos/sin (row constant in group),
//            4 bilinear taps from the LDS tile, one global_store_b128 to the
//            row-rolled destination, scaled by 1/255.
// ----------------------------------------------------------------------------
__global__ __launch_bounds__(256) void logpolar_patch_kernel(
    const float* __restrict__ img,      // [B,1,256,256]
    const float* __restrict__ kpLoc,    // [B,2]
    const float* __restrict__ scaling,  // [B]
    const float* __restrict__ rotation, // [B]
    float* __restrict__ out)            // [B,1,64,64]
{
    __shared__ __align__(16) float tile[TILE * TILE];
    __shared__ __align__(16) float rp[RES];   // radius (pixels), per column j
    __shared__ float cs[RES];                 // cos(theta_i), per row i
    __shared__ float sn[RES];                 // sin(theta_i), per row i

    const int b   = blockIdx.x;
    const int tid = threadIdx.x;

    const float kpx  = kpLoc[2 * b + 0];
    const float kpy  = kpLoc[2 * b + 1];
    const float maxR = 6.0f * scaling[b];      // 0.5 * SIFTSCALE * scaling
    const float rot  = rotation[b];

    // keypoint center in pixel coords (align_corners=False mapping)
    const float cx = kpx * 128.0f + 127.5f;    // ((x+1)*W - 1) * 0.5
    const float cy = kpy * 128.0f + 127.5f;

    // tile origin: sample radius <= ~2.1 px (+1 bilinear) around center, which
    // sits >= ~50 px from any border, so a 16x16 tile always covers it.
    // ox kept multiple of 4 so the B128 global loads stay 16B-aligned
    // (clamp bounds 0 and 240 are both multiples of 4).
    int ox = (((int)floorf(cx)) - 6) & ~3;
    int oy = ((int)floorf(cy)) - 6;
    ox = max(0, min(ox, IMG_W - TILE));
    oy = max(0, min(oy, IMG_H - TILE));

    const float* gbase = img + (size_t)b * (IMG_H * IMG_W);

    if (tid < RES) {
        // radius per column j = tid:  normGrid = (2j+1)/128
        const float norm = (float)(2 * tid + 1) * (1.0f / 128.0f);
        const float rs_  = expf(norm * logf(maxR));                 // maxR**normGrid
        const float r_s  = (rs_ - 1.0f) / (maxR - 1.0f) * (2.0f * maxR / 1500.0f);
        rp[tid] = r_s * 128.0f;                                     // normalized->pixels

        // angle per row i = tid: t = (2i+1)*pi/64
        const float t = (float)(2 * tid + 1) * (PI_F / 64.0f);
        float s, c;
        sincosf(t, &s, &c);
        cs[tid] = c;
        sn[tid] = s;

        // stage the 16x16 tile: 16 rows x 4 segments of 16B each (64 lanes)
        const int row = tid >> 2;
        const int seg = tid & 3;
        const float* gp = gbase + (size_t)(oy + row) * IMG_W + ox + seg * 4;
        float*       lp = &tile[row * TILE + seg * 4];
#if USE_ASYNC_LDS
        __builtin_amdgcn_global_load_async_to_lds_b128(
            (gv4i_t*)(uintptr_t)gp,
            (lv4i_t*)(uint32_t)(uintptr_t)lp,
            /*offset=*/0, /*cpol=*/0);
#else
        *(float4*)lp = *(const float4*)gp;
#endif
    }

#if USE_ASYNC_LDS
    wait_asynccnt0();
#endif
    __syncthreads();

    // circular roll along rows: out[i] = patch[(i-n) mod 64]  =>  row r -> (r+n)&63
    int n = (int)rintf(rot * (32.0f / PI_F));   // RNE matches jnp.round
    n = ((n % RES) + RES) % RES;

    const float fx0 = cx - (float)ox;           // tile-local center
    const float fy0 = cy - (float)oy;
    float* obase = out + (size_t)b * (RES * RES);

#pragma unroll
    for (int k = 0; k < 4; ++k) {
        const int g  = tid + (k << 8);          // group id 0..1023
        const int i  = g >> 4;                  // row (angle) 0..63
        const int j0 = (g & 15) << 2;           // col base 0,4,...,60

        const float c = cs[i];
        const float s = sn[i];
        const float4 rv = *(const float4*)&rp[j0];   // ds_load_b128

        float4 res;
        float* resp = (float*)&res;
        const float* rvp = (const float*)&rv;
#pragma unroll
        for (int m = 0; m < 4; ++m) {
            const float r  = rvp[m];
            const float ix = r * c + fx0;       // tile-local sample coords
            const float iy = r * s + fy0;

            const float xf = floorf(ix), yf = floorf(iy);
            const float wx1 = ix - xf, wy1 = iy - yf;
            const float wx0 = 1.0f - wx1, wy0 = 1.0f - wy1;

            int x0 = (int)xf, y0 = (int)yf;
            x0 = max(0, min(x0, TILE - 2));     // defensive; interior by design
            y0 = max(0, min(y0, TILE - 2));

            const float* t0 = &tile[y0 * TILE + x0];
            const float v = wy0 * (wx0 * t0[0]    + wx1 * t0[1])
                          + wy1 * (wx0 * t0[TILE] + wx1 * t0[TILE + 1]);
            resp[m] = v * (1.0f / 255.0f);
        }

        // one 16B store per group; rows permuted by the roll, columns contiguous
        *(float4*)&obase[(((i + n) & 63) << 6) | j0] = res;
    }
}

extern "C" void kernel_launch(void* const* d_in, const int* in_sizes, int n_in,
                              void* d_out, int out_size, void* d_ws, size_t ws_size,
                              hipStream_t stream) {
    const float* img      = (const float*)d_in[0];
    const float* kpLoc    = (const float*)d_in[1];
    const float* scaling  = (const float*)d_in[2];
    const float* rotation = (const float*)d_in[3];
    float* out            = (float*)d_out;

    const int B = in_sizes[2];  // scaling has B elements
    hipLaunchKernelGGL(logpolar_patch_kernel, dim3(B), dim3(256), 0, stream,
                       img, kpLoc, scaling, rotation, out);
}